// MatrixModel_66331474920016
// MI455X (gfx1250) — compile-verified
//
#include <hip/hip_runtime.h>

#define NUM_LABELS 10

typedef float v2f __attribute__((ext_vector_type(2)));
typedef float v8f __attribute__((ext_vector_type(8)));

// Cross-lane gather via LDS permute hardware (no LDS memory used, wave32).
static __device__ __forceinline__ float lane_gather(float v, int srcLane) {
    return __int_as_float(__builtin_amdgcn_ds_bpermute(srcLane << 2, __float_as_int(v)));
}

__global__ void __launch_bounds__(256)
lidar_project_wmma_kernel(const float* __restrict__ lidar,
                          const float* __restrict__ Tr,
                          const float* __restrict__ P,
                          const int*   __restrict__ label,
                          const int*   __restrict__ pH,
                          const int*   __restrict__ pW,
                          float*       __restrict__ out,
                          int N)
{
    const int lane  = threadIdx.x & 31;
    const int m     = lane & 15;      // A-matrix row this lane feeds
    const int khalf = lane >> 4;      // 0 -> K=0,1 ; 1 -> K=2,3
    const int j0    = khalf * 2;
    const int j1    = j0 + 1;

    // ---- Fused 4x4 transform (loop invariant) -----------------------------
    // Row0    = Tr[2][:]                       (depth = cam_z)
    // Row1..3 = P[i][:3] @ Tr  (+ P[i][3] in last column)   (hom = P @ [cam;1])
    // Each lane only materializes the two A-entries it owns (no scratch arrays).
    auto homRow = [&](int i, int j) -> float {
        float s = (j == 3) ? P[i * 4 + 3] : 0.0f;
        s += P[i * 4 + 0] * Tr[0 * 4 + j];
        s += P[i * 4 + 1] * Tr[1 * 4 + j];
        s += P[i * 4 + 2] * Tr[2 * 4 + j];
        return s;
    };
    float r0j0 = Tr[2 * 4 + j0], r0j1 = Tr[2 * 4 + j1];
    float r1j0 = homRow(0, j0),  r1j1 = homRow(0, j1);
    float r2j0 = homRow(1, j0),  r2j1 = homRow(1, j1);
    float r3j0 = homRow(2, j0),  r3j1 = homRow(2, j1);

    v2f a;
    a.x = (m == 0) ? r0j0 : (m == 1) ? r1j0 : (m == 2) ? r2j0 : (m == 3) ? r3j0 : 0.0f;
    a.y = (m == 0) ? r0j1 : (m == 1) ? r1j1 : (m == 2) ? r2j1 : (m == 3) ? r3j1 : 0.0f;

    const float wlim = (float)(*pW - 1);
    const float hlim = (float)(*pH - 1);

    const int wavesPerBlock = blockDim.x >> 5;
    const int waveId        = blockIdx.x * wavesPerBlock + (threadIdx.x >> 5);
    const int totalWaves    = gridDim.x * wavesPerBlock;
    const bool lo           = lane < 16;

    for (int base = waveId * 32; base < N; base += totalWaves * 32) {
        const int p  = base + lane;
        const int pc = (p < N) ? p : (N - 1);   // clamp loads; stores masked later

        const float x   = lidar[(size_t)pc * 3 + 0];
        const float y   = lidar[(size_t)pc * 3 + 1];
        const float z   = lidar[(size_t)pc * 3 + 2];
        const int   lab = label[pc];

        // Swap halves so both B operands can be built branchlessly.
        const float sx = lane_gather(x, lane ^ 16);
        const float sy = lane_gather(y, lane ^ 16);
        const float sz = lane_gather(z, lane ^ 16);

        // B (4x16 f32): lanes 0-15 carry K=0,1 (x,y); lanes 16-31 carry K=2,3 (z,1)
        v2f b1; b1.x = lo ? x  : sz; b1.y = lo ? y  : 1.0f;   // points base..base+15
        v2f b2; b2.x = lo ? sx : z;  b2.y = lo ? sy : 1.0f;   // points base+16..base+31

        v8f c = {0.f, 0.f, 0.f, 0.f, 0.f, 0.f, 0.f, 0.f};
        // D rows 0..3 = (depth, hom_x, hom_y, hom_z) per point-column.
        v8f d1 = __builtin_amdgcn_wmma_f32_16x16x4_f32(false, a, false, b1,
                                                       (short)0, c, false, false);
        v8f d2 = __builtin_amdgcn_wmma_f32_16x16x4_f32(false, a, false, b2,
                                                       (short)0, c, false, false);

        // Lane n<16 owns point base+n (from d1); lane n>=16 owns point base+n,
        // whose results live in lane n-16 of d2.
        const float t0 = lane_gather(d2[0], lane & 15);
        const float t1 = lane_gather(d2[1], lane & 15);
        const float t2 = lane_gather(d2[2], lane & 15);
        const float t3 = lane_gather(d2[3], lane & 15);
        const float depth = lo ? d1[0] : t0;
        const float hx    = lo ? d1[1] : t1;
        const float hy    = lo ? d1[2] : t2;
        const float hz    = lo ? d1[3] : t3;

        // Fast reciprocal (v_rcp_f32, ~1 ulp) instead of the ~12-instr IEEE
        // div expansion: this kernel is on the compute/memory balance point,
        // and the quotient only feeds image coords + range compares.
        const float rz = __builtin_amdgcn_rcpf(hz);
        const float u  = hx * rz;
        const float v  = hy * rz;
        // NaN compares are false -> fov false -> zeros, matching jnp.where.
        const bool fov = (u < wlim) & (u >= 0.0f) & (v < hlim) & (v >= 0.0f) & (x > 2.0f);

        if (p < N) {   // tail mask only around stores; EXEC full at the WMMAs
            #pragma unroll
            for (int l = 0; l < NUM_LABELS; ++l) {
                const bool  sel = fov & (lab == l);
                const float o0  = sel ? u     : 0.0f;
                const float o1  = sel ? v     : 0.0f;
                const float o2  = sel ? depth : 0.0f;
                float* dst = out + ((size_t)l * (size_t)N + (size_t)p) * 3;
                // Output (252 MB) is write-once streaming > L2 (192 MB): NT stores.
                __builtin_nontemporal_store(o0, dst + 0);
                __builtin_nontemporal_store(o1, dst + 1);
                __builtin_nontemporal_store(o2, dst + 2);
            }
        }
    }
}

extern "C" void kernel_launch(void* const* d_in, const int* in_sizes, int n_in,
                              void* d_out, int out_size, void* d_ws, size_t ws_size,
                              hipStream_t stream) {
    const float* lidar = (const float*)d_in[0];
    const float* Tr    = (const float*)d_in[1];
    const float* P     = (const float*)d_in[2];
    const int*   label = (const int*)d_in[3];
    const int*   pH    = (const int*)d_in[4];
    const int*   pW    = (const int*)d_in[5];
    float*       out   = (float*)d_out;

    const int N = in_sizes[0] / 3;

    const int threads = 256;              // 8 waves/block on wave32
    int blocks = (N + threads - 1) / threads;
    if (blocks < 1) blocks = 1;

    lidar_project_wmma_kernel<<<blocks, threads, 0, stream>>>(
        lidar, Tr, P, label, pH, pW, out, N);
}